// Cubic_71201967833506
// MI455X (gfx1250) — compile-verified
//
#include <hip/hip_runtime.h>

typedef float v2f __attribute__((ext_vector_type(2)));
typedef float v8f __attribute__((ext_vector_type(8)));

// ---------------------------------------------------------------------------
// Interior weight table WT[n][k]: weight of input tap (base+k) for output
// (16*tile + n), scale 4, Keys a=-0.5. Exact binary fractions; each row sums
// to 1 (interior => no renormalization). Same table serves horizontal (as
// stage-1 B) and vertical (as stage-2 A) because the geometry is identical.
// Phase vectors (tf = frac of src coord):
//   p0 tf=.625: -45/1024, 399/1024, 745/1024, -75/1024
//   p1 tf=.875:  -7/1024,  93/1024, 987/1024, -49/1024
//   p2 tf=.125: reverse of p1;  p3 tf=.375: reverse of p0
// ---------------------------------------------------------------------------
__constant__ __attribute__((aligned(16))) float WT[16][8] = {
  {-0.0439453125f, 0.3896484375f, 0.7275390625f,-0.0732421875f, 0.f, 0.f, 0.f, 0.f},
  {-0.0068359375f, 0.0908203125f, 0.9638671875f,-0.0478515625f, 0.f, 0.f, 0.f, 0.f},
  {0.f,-0.0478515625f, 0.9638671875f, 0.0908203125f,-0.0068359375f, 0.f, 0.f, 0.f},
  {0.f,-0.0732421875f, 0.7275390625f, 0.3896484375f,-0.0439453125f, 0.f, 0.f, 0.f},
  {0.f,-0.0439453125f, 0.3896484375f, 0.7275390625f,-0.0732421875f, 0.f, 0.f, 0.f},
  {0.f,-0.0068359375f, 0.0908203125f, 0.9638671875f,-0.0478515625f, 0.f, 0.f, 0.f},
  {0.f, 0.f,-0.0478515625f, 0.9638671875f, 0.0908203125f,-0.0068359375f, 0.f, 0.f},
  {0.f, 0.f,-0.0732421875f, 0.7275390625f, 0.3896484375f,-0.0439453125f, 0.f, 0.f},
  {0.f, 0.f,-0.0439453125f, 0.3896484375f, 0.7275390625f,-0.0732421875f, 0.f, 0.f},
  {0.f, 0.f,-0.0068359375f, 0.0908203125f, 0.9638671875f,-0.0478515625f, 0.f, 0.f},
  {0.f, 0.f, 0.f,-0.0478515625f, 0.9638671875f, 0.0908203125f,-0.0068359375f, 0.f},
  {0.f, 0.f, 0.f,-0.0732421875f, 0.7275390625f, 0.3896484375f,-0.0439453125f, 0.f},
  {0.f, 0.f, 0.f,-0.0439453125f, 0.3896484375f, 0.7275390625f,-0.0732421875f, 0.f},
  {0.f, 0.f, 0.f,-0.0068359375f, 0.0908203125f, 0.9638671875f,-0.0478515625f, 0.f},
  {0.f, 0.f, 0.f, 0.f,-0.0478515625f, 0.9638671875f, 0.0908203125f,-0.0068359375f},
  {0.f, 0.f, 0.f, 0.f,-0.0732421875f, 0.7275390625f, 0.3896484375f,-0.0439453125f},
};

// Branchless Keys cubic (a=-0.5): avoids EXEC-mask branching.
__device__ __forceinline__ float keys_w(float s) {
    s = fabsf(s);
    const float wn = (1.5f * s - 2.5f) * s * s + 1.f;          // |s| < 1
    const float wf = ((-0.5f * s + 2.5f) * s - 4.f) * s + 2.f; // 1 <= |s| < 2
    float w = (s < 1.f) ? wn : wf;
    return (s < 2.f) ? w : 0.f;
}

// Boundary weight: out-of-range taps dropped, remainder renormalized (jax).
__device__ __forceinline__ float bicubic_w(int o, int t, int n_in) {
    const int q = o >> 2, p = o & 3;
    const float x = (float)q + 0.125f * (float)(2 * p - 3);
    const int f = (p < 2) ? (q - 1) : q;
    float sum = 0.f, wt = 0.f;
#pragma unroll
    for (int k = 0; k < 4; ++k) {
        const int tk = f - 1 + k;
        float w = keys_w(x - (float)tk);
        w = (tk >= 0 && tk < n_in) ? w : 0.f;
        sum += w;
        if (tk == t) wt = w;
    }
    return wt / sum;
}

__device__ __forceinline__ int iclamp(int v, int lo, int hi) {
    return v < lo ? lo : (v > hi ? hi : v);
}

// One wave32 -> one 16x16 output tile; 8 waves per 256-thread block.
__global__ __launch_bounds__(256)
void Cubic4x_kernel(const float* __restrict__ in, float* __restrict__ out,
                    int n_tiles) {
    __shared__ float lds[8][8][16];  // per-wave T tile (stage-1 rows 0..7)

    const int lane = threadIdx.x & 31;
    const int wv   = threadIdx.x >> 5;
    const int tile = blockIdx.x * 8 + wv;

    const int tx  = tile & 63;
    const int ty  = (tile >> 6) & 63;
    const int img = tile >> 12;

    const int C  = tx * 16, R = ty * 16;
    const int cb = tx * 4 - 2, rb = ty * 4 - 2;

    const int n  = lane & 15;  // A-row / B-col / D-col for this lane
    const int hi = lane >> 4;  // K half: {0,1} vs {2,3}
    const int k0 = 2 * hi;     // first K index of each chunk (chunk1 adds 4)

    const float* __restrict__ src = in + (size_t)img * (256 * 256);

    v2f A1[2];  // stage-1 A (input patch), chunks kb=0,4
    v2f BH[2];  // stage-1 B (horizontal weights)
    v2f AV[2];  // stage-2 A (vertical weights)

    const bool interior = (tx > 0) & (tx < 63) & (ty > 0) & (ty < 63);
    if (interior) {
        // Rows 8..15 of A are never consumed (T rows 8..15 unread): duplicate
        // rows 0..7 so all addresses stay in-bounds without clamping.
        const float* rowp = src + (rb + (n & 7)) * 256 + cb;
        A1[0] = *(const v2f*)(rowp + k0);        // cols cb+k0, cb+k0+1
        A1[1] = *(const v2f*)(rowp + 4 + k0);    // cols cb+k0+4, cb+k0+5
        const v2f* wrow = (const v2f*)(&WT[n][0]);
        BH[0] = wrow[hi];
        BH[1] = wrow[hi + 2];
        AV[0] = BH[0];
        AV[1] = BH[1];
    } else {
        const int r  = iclamp(rb + (n & 7), 0, 255);
        const int c0 = iclamp(cb + k0,     0, 255);
        const int c1 = iclamp(cb + k0 + 1, 0, 255);
        const int c2 = iclamp(cb + k0 + 4, 0, 255);
        const int c3 = iclamp(cb + k0 + 5, 0, 255);
        const float* rowp = src + r * 256;
        A1[0][0] = rowp[c0];  A1[0][1] = rowp[c1];
        A1[1][0] = rowp[c2];  A1[1][1] = rowp[c3];
        BH[0][0] = bicubic_w(C + n, cb + k0,     256);
        BH[0][1] = bicubic_w(C + n, cb + k0 + 1, 256);
        BH[1][0] = bicubic_w(C + n, cb + k0 + 4, 256);
        BH[1][1] = bicubic_w(C + n, cb + k0 + 5, 256);
        AV[0][0] = bicubic_w(R + n, rb + k0,     256);
        AV[0][1] = bicubic_w(R + n, rb + k0 + 1, 256);
        AV[1][0] = bicubic_w(R + n, rb + k0 + 4, 256);
        AV[1][1] = bicubic_w(R + n, rb + k0 + 5, 256);
    }

    // -------- Stage 1: horizontal  T(16x16) = Inpad(16x8) x H(8x16) --------
    v8f t = {};
    t = __builtin_amdgcn_wmma_f32_16x16x4_f32(false, A1[0], false, BH[0],
                                              (short)0, t, false, false);
    t = __builtin_amdgcn_wmma_f32_16x16x4_f32(false, A1[1], false, BH[1],
                                              (short)0, t, false, false);

    // D layout -> LDS: vgpr r holds T[r][n] in lanes 0..15 (rows 8..15 unused)
    if (hi == 0) {
#pragma unroll
        for (int r = 0; r < 8; ++r) lds[wv][r][n] = t[r];
    }
    __syncthreads();

    // -------- Stage 2: vertical  Out(16x16) = V(16x8) x T(8x16) ------------
    v2f b0, b1;
    b0[0] = lds[wv][k0][n];      b0[1] = lds[wv][k0 + 1][n];
    b1[0] = lds[wv][k0 + 4][n];  b1[1] = lds[wv][k0 + 5][n];

    v8f acc = {};
    acc = __builtin_amdgcn_wmma_f32_16x16x4_f32(false, AV[0], false, b0,
                                                (short)0, acc, false, false);
    acc = __builtin_amdgcn_wmma_f32_16x16x4_f32(false, AV[1], false, b1,
                                                (short)0, acc, false, false);

    // D -> global: vgpr r = row R + r + 8*hi, col C + n. Output streams past
    // cache (201 MB): non-temporal stores.
    float* __restrict__ dst = out + (size_t)img * (1024 * 1024);
#pragma unroll
    for (int r = 0; r < 8; ++r) {
        const int orow = R + r + 8 * hi;
        __builtin_nontemporal_store(acc[r], &dst[(size_t)orow * 1024 + C + n]);
    }
}

extern "C" void kernel_launch(void* const* d_in, const int* in_sizes, int n_in,
                              void* d_out, int out_size, void* d_ws, size_t ws_size,
                              hipStream_t stream) {
    const float* x = (const float*)d_in[0];
    float* out = (float*)d_out;

    const int images  = in_sizes[0] / (256 * 256);   // 48
    const int n_tiles = images * 64 * 64;            // 196608 (multiple of 8)
    const int blocks  = n_tiles / 8;                 // 8 waves (tiles) / block

    hipLaunchKernelGGL(Cubic4x_kernel, dim3(blocks), dim3(256), 0, stream,
                       x, out, n_tiles);
}